// GCN_850403525401
// MI455X (gfx1250) — compile-verified
//
#include <hip/hip_runtime.h>
#include <hip/hip_bf16.h>
#include <math.h>

typedef __attribute__((ext_vector_type(2))) float v2f;
typedef __attribute__((ext_vector_type(8))) float v8f;

#define C_IN  128
#define C_HID 64
#define C_OUT 16

// V_WMMA_F32_16X16X4_F32: D(16x16 f32) = A(16x4 f32) x B(4x16 f32) + C
// Per-lane fragments (wave32):
//   A: lane l holds rows m=l&15, a[0]=A[m][kb+2*(l>>4)], a[1]=A[m][kb+2*(l>>4)+1]
//   B: lane l holds col n=l&15,  b[0]=B[kb+2*(l>>4)][n], b[1]=B[kb+2*(l>>4)+1][n]
//   C/D: VGPR r -> row (r + 8*(l>>4)), col l&15
static __device__ __forceinline__ v8f wmma_f32x4(v2f a, v2f b, v8f c) {
  return __builtin_amdgcn_wmma_f32_16x16x4_f32(false, a, false, b, (short)0, c,
                                               false, false);
}

static __device__ __forceinline__ void atom_add_f32(float* p, float v) {
  __hip_atomic_fetch_add(p, v, __ATOMIC_RELAXED, __HIP_MEMORY_SCOPE_AGENT);
}

// ---------------------------------------------------------------------------
// Layer-1 dense part: y_rel = x @ w_rel^T   (N x 64)
//                     h     = x @ w_root^T + b1   (N x 64, scatter target)
// One wave computes a 16-row x 64-col tile of both outputs (8 accumulators).
// Both weight matrices are staged into LDS once per workgroup; B fragments
// then come from ds_load_b64 while the next A fragment is prefetched from
// global, so WMMAs no longer serialize on full VMEM latency.
// ---------------------------------------------------------------------------
__global__ void gcn_gemm1(const float* __restrict__ x,
                          const float* __restrict__ w_rel,
                          const float* __restrict__ w_root,
                          const float* __restrict__ bias,
                          float* __restrict__ y_rel,
                          float* __restrict__ h,
                          int n_rows, int n_tiles) {
  __shared__ float lw[2 * C_HID * C_IN];          // 64 KB: w_rel | w_root
  {
    const int t = threadIdx.x, nthr = blockDim.x;
    const float4* g0 = (const float4*)w_rel;
    const float4* g1 = (const float4*)w_root;
    float4* s0 = (float4*)lw;
    float4* s1 = (float4*)(lw + C_HID * C_IN);
    const int n4 = (C_HID * C_IN) / 4;            // 2048 float4s each
    for (int i = t; i < n4; i += nthr) {
      s0[i] = g0[i];
      s1[i] = g1[i];
    }
  }
  __syncthreads();

  const int lane = threadIdx.x & 31;
  const int wave = threadIdx.x >> 5;
  const int tile = blockIdx.x * (blockDim.x >> 5) + wave;
  if (tile >= n_tiles) return;
  const int lo = lane & 15, hi = lane >> 4;
  const int rowbase = tile * 16;

  int arow = rowbase + lo;
  if (arow >= n_rows) arow = n_rows - 1;          // clamp (stores are guarded)
  const float* xr = x + (size_t)arow * C_IN;
  const float* wr = lw;
  const float* wo = lw + C_HID * C_IN;

  v8f acc_rel[4] = {};
  v8f acc_root[4] = {};

  v2f a = *(const v2f*)(xr + 2 * hi);             // prefetch k-step 0
  for (int kb = 0; kb < C_IN; kb += 4) {
    const int k0 = kb + 2 * hi;
    const v2f a_cur = a;
    if (kb + 4 < C_IN)                            // prefetch next k-step
      a = *(const v2f*)(xr + k0 + 4);
#pragma unroll
    for (int c = 0; c < 4; ++c) {
      const int col = c * 16 + lo;                // output channel
      const v2f br = *(const v2f*)(wr + col * C_IN + k0);   // ds_load_b64
      const v2f bo = *(const v2f*)(wo + col * C_IN + k0);   // ds_load_b64
      acc_rel[c]  = wmma_f32x4(a_cur, br, acc_rel[c]);
      acc_root[c] = wmma_f32x4(a_cur, bo, acc_root[c]);
    }
  }

#pragma unroll
  for (int c = 0; c < 4; ++c) {
    const int col = c * 16 + lo;
    const float bv = bias[col];
#pragma unroll
    for (int r = 0; r < 8; ++r) {
      const int row = rowbase + r + 8 * hi;
      if (row < n_rows) {
        y_rel[(size_t)row * C_HID + col] = acc_rel[c][r];
        h[(size_t)row * C_HID + col]     = acc_root[c][r] + bv;
      }
    }
  }
}

// ---------------------------------------------------------------------------
// Edge scatter, layer 1: h[dst] += y_rel[src]  (64 floats/edge, 16 lanes x f4)
// ---------------------------------------------------------------------------
__global__ void gcn_scatter1(const long long* __restrict__ src,
                             const long long* __restrict__ dst,
                             const float* __restrict__ y_rel,
                             float* __restrict__ h, int n_edges) {
  const int gid = blockIdx.x * blockDim.x + threadIdx.x;
  const int e = gid >> 4;
  const int part = gid & 15;
  if (e >= n_edges) return;
  const long long s = src[e];
  const long long d = dst[e];
  const float4 v = *(const float4*)(y_rel + (size_t)s * C_HID + part * 4);
  float* p = h + (size_t)d * C_HID + part * 4;
  atom_add_f32(p + 0, v.x);
  atom_add_f32(p + 1, v.y);
  atom_add_f32(p + 2, v.z);
  atom_add_f32(p + 3, v.w);
}

// ---------------------------------------------------------------------------
// Layer-2 dense part on ReLU(h): y2 = relu(h) @ w2_rel^T   (N x 16)
//                                o  = relu(h) @ w2_root^T + b2  -> d_out init
// ---------------------------------------------------------------------------
__global__ void gcn_gemm2(const float* __restrict__ h,
                          const float* __restrict__ w_rel,
                          const float* __restrict__ w_root,
                          const float* __restrict__ bias,
                          float* __restrict__ y2,
                          float* __restrict__ o,
                          int n_rows, int n_tiles) {
  __shared__ float lw[2 * C_OUT * C_HID];         // 8 KB: w2_rel | w2_root
  {
    const int t = threadIdx.x, nthr = blockDim.x;
    const float4* g0 = (const float4*)w_rel;
    const float4* g1 = (const float4*)w_root;
    float4* s0 = (float4*)lw;
    float4* s1 = (float4*)(lw + C_OUT * C_HID);
    const int n4 = (C_OUT * C_HID) / 4;           // 256 float4s each
    for (int i = t; i < n4; i += nthr) {
      s0[i] = g0[i];
      s1[i] = g1[i];
    }
  }
  __syncthreads();

  const int lane = threadIdx.x & 31;
  const int wave = threadIdx.x >> 5;
  const int tile = blockIdx.x * (blockDim.x >> 5) + wave;
  if (tile >= n_tiles) return;
  const int lo = lane & 15, hi = lane >> 4;
  const int rowbase = tile * 16;

  int arow = rowbase + lo;
  if (arow >= n_rows) arow = n_rows - 1;
  const float* hr = h + (size_t)arow * C_HID;
  const float* wr = lw;
  const float* wo = lw + C_OUT * C_HID;

  v8f acc_rel = {};
  v8f acc_root = {};

  v2f a = *(const v2f*)(hr + 2 * hi);             // prefetch k-step 0
  for (int kb = 0; kb < C_HID; kb += 4) {
    const int k0 = kb + 2 * hi;
    v2f a_cur = a;
    if (kb + 4 < C_HID)
      a = *(const v2f*)(hr + k0 + 4);
    a_cur.x = fmaxf(a_cur.x, 0.0f);               // fused ReLU on A fragment
    a_cur.y = fmaxf(a_cur.y, 0.0f);
    const v2f br = *(const v2f*)(wr + lo * C_HID + k0);     // ds_load_b64
    const v2f bo = *(const v2f*)(wo + lo * C_HID + k0);     // ds_load_b64
    acc_rel  = wmma_f32x4(a_cur, br, acc_rel);
    acc_root = wmma_f32x4(a_cur, bo, acc_root);
  }

  const float bv = bias[lo];
#pragma unroll
  for (int r = 0; r < 8; ++r) {
    const int row = rowbase + r + 8 * hi;
    if (row < n_rows) {
      y2[(size_t)row * C_OUT + lo] = acc_rel[r];
      o[(size_t)row * C_OUT + lo]  = acc_root[r] + bv;
    }
  }
}

// ---------------------------------------------------------------------------
// Edge scatter, layer 2: o[dst] += y2[src]  (16 floats/edge, 4 lanes x f4)
// ---------------------------------------------------------------------------
__global__ void gcn_scatter2(const long long* __restrict__ src,
                             const long long* __restrict__ dst,
                             const float* __restrict__ y2,
                             float* __restrict__ o, int n_edges) {
  const int gid = blockIdx.x * blockDim.x + threadIdx.x;
  const int e = gid >> 2;
  const int part = gid & 3;
  if (e >= n_edges) return;
  const long long s = src[e];
  const long long d = dst[e];
  const float4 v = *(const float4*)(y2 + (size_t)s * C_OUT + part * 4);
  float* p = o + (size_t)d * C_OUT + part * 4;
  atom_add_f32(p + 0, v.x);
  atom_add_f32(p + 1, v.y);
  atom_add_f32(p + 2, v.z);
  atom_add_f32(p + 3, v.w);
}

// ---------------------------------------------------------------------------
// In-place log_softmax over the 16 columns of each row of o (= d_out).
// ---------------------------------------------------------------------------
__global__ void gcn_logsoftmax(float* __restrict__ o, int n_rows) {
  const int i = blockIdx.x * blockDim.x + threadIdx.x;
  if (i >= n_rows) return;
  float* row = o + (size_t)i * C_OUT;
  float v[C_OUT];
  const float4* r4 = (const float4*)row;
#pragma unroll
  for (int q = 0; q < 4; ++q) {
    const float4 t = r4[q];
    v[q * 4 + 0] = t.x; v[q * 4 + 1] = t.y;
    v[q * 4 + 2] = t.z; v[q * 4 + 3] = t.w;
  }
  float m = v[0];
#pragma unroll
  for (int c = 1; c < C_OUT; ++c) m = fmaxf(m, v[c]);
  float s = 0.0f;
#pragma unroll
  for (int c = 0; c < C_OUT; ++c) s += __expf(v[c] - m);
  const float lse = m + __logf(s);
#pragma unroll
  for (int c = 0; c < C_OUT; ++c) row[c] = v[c] - lse;
}

extern "C" void kernel_launch(void* const* d_in, const int* in_sizes, int n_in,
                              void* d_out, int out_size, void* d_ws, size_t ws_size,
                              hipStream_t stream) {
  const float*     x       = (const float*)d_in[0];
  const long long* ei      = (const long long*)d_in[1];   // int64 [2, E]
  const float*     w1_rel  = (const float*)d_in[2];
  const float*     w1_root = (const float*)d_in[3];
  const float*     b1      = (const float*)d_in[4];
  const float*     w2_rel  = (const float*)d_in[5];
  const float*     w2_root = (const float*)d_in[6];
  const float*     b2      = (const float*)d_in[7];
  float*           out     = (float*)d_out;

  const int N = in_sizes[0] / C_IN;
  const int E = in_sizes[1] / 2;
  const long long* src = ei;
  const long long* dst = ei + E;

  // Workspace: y1 (N*64) | h (N*64) | y2 (N*16)  -> 57.6 MB for N=100000
  float* y1 = (float*)d_ws;
  float* h  = y1 + (size_t)N * C_HID;
  float* y2 = h + (size_t)N * C_HID;

  const int row_tiles = (N + 15) / 16;
  const int waves_per_block = 8;                       // 256 threads
  const int gemm_blocks = (row_tiles + waves_per_block - 1) / waves_per_block;

  gcn_gemm1<<<gemm_blocks, waves_per_block * 32, 0, stream>>>(
      x, w1_rel, w1_root, b1, y1, h, N, row_tiles);

  {
    const long long work = (long long)E * 16;
    const int blocks = (int)((work + 255) / 256);
    gcn_scatter1<<<blocks, 256, 0, stream>>>(src, dst, y1, h, E);
  }

  gcn_gemm2<<<gemm_blocks, waves_per_block * 32, 0, stream>>>(
      h, w2_rel, w2_root, b2, y2, out, N, row_tiles);

  {
    const long long work = (long long)E * 4;
    const int blocks = (int)((work + 255) / 256);
    gcn_scatter2<<<blocks, 256, 0, stream>>>(src, dst, y2, out, E);
  }

  gcn_logsoftmax<<<(N + 255) / 256, 256, 0, stream>>>(out, N);
}